// QValueAgent_11639361372464
// MI455X (gfx1250) — compile-verified
//
#include <hip/hip_runtime.h>
#include <cstdint>

#define N_SESS          8192
#define N_TRIALS        1024
#define WAVES_PER_BLOCK 4
#define LANES           32
#define TPL             (N_TRIALS / LANES)          // 32 trials per lane
#define SESS_FLOATS     (N_TRIALS * 3)              // 3072 floats / session
#define SESS_BYTES      (SESS_FLOATS * 4)           // 12288 B / session
#define ASYNC_OPS       (SESS_BYTES / (LANES * 16)) // 24 b128 async loads / wave

__global__ __launch_bounds__(WAVES_PER_BLOCK * LANES)
void qvalue_scan_kernel(const float* __restrict__ inp,
                        const float* __restrict__ alpha_raw,
                        const float* __restrict__ kvec,
                        float* __restrict__ out)
{
    // 4 waves x 12KB input staging = 48 KB LDS per workgroup
    __shared__ float s_in[WAVES_PER_BLOCK][SESS_FLOATS];

    const int lane = threadIdx.x & (LANES - 1);
    const int wave = threadIdx.x >> 5;
    const int sess = blockIdx.x * WAVES_PER_BLOCK + wave;

    // alphas = sigmoid(alpha_raw); q_i = alpha_i * k_i  (uniform, scalar loads)
    const float a0 = 1.0f / (1.0f + __expf(-alpha_raw[0]));
    const float a1 = 1.0f / (1.0f + __expf(-alpha_raw[1]));
    const float a2 = 1.0f / (1.0f + __expf(-alpha_raw[2]));
    const float a3 = 1.0f / (1.0f + __expf(-alpha_raw[3]));
    const float q0 = a0 * kvec[0];
    const float q1 = a1 * kvec[1];
    const float q2 = a2 * kvec[2];
    const float q3 = a3 * kvec[3];

    // ---- Phase A: async-stage this session's 12KB of input into LDS ----
    // GVS addressing: SGPR64 base + 32-bit VGPR offset (max offset ~100MB < 2^31).
    const uint32_t sessByte = (uint32_t)sess * (uint32_t)SESS_BYTES;
    const uint32_t ldsBase  = (uint32_t)(uintptr_t)(&s_in[wave][0]);
#pragma unroll
    for (int i = 0; i < ASYNC_OPS; ++i) {
        uint32_t goff = sessByte + (uint32_t)(i * LANES + lane) * 16u;
        uint32_t loff = ldsBase  + (uint32_t)(i * LANES + lane) * 16u;
        asm volatile("global_load_async_to_lds_b128 %0, %1, %2"
                     :: "v"(loff), "v"(goff), "s"(inp)
                     : "memory");
    }
    asm volatile("s_wait_asynccnt 0x0" ::: "memory");
    // Each wave only touches its own LDS region -> no workgroup barrier needed.

    // ---- Phase B: compose affine maps over this lane's 32 trials ----
    // Step: Q' = m*Q + c with m = 1 - A, c = B.
    const float* __restrict__ my = &s_in[wave][lane * TPL * 3];
    float Ml = 1.0f, Cl = 0.0f, Mr = 1.0f, Cr = 0.0f;
#pragma unroll
    for (int t = 0; t < TPL; ++t) {
        const float cl = my[t * 3 + 0];
        const float cr = my[t * 3 + 1];
        const float o  = my[t * 3 + 2];
        const float u  = fmaf(o, a0 - a1, a1);   // o?a0:a1
        const float v  = fmaf(o, a2 - a3, a3);   // o?a2:a3
        const float uk = fmaf(o, q0 - q1, q1);
        const float vk = fmaf(o, q2 - q3, q3);
        const float ml = 1.0f - (fmaf(cl, u,  cr * v));   // 1 - A_l
        const float bl =         fmaf(cl, uk, cr * vk);   // B_l
        const float mr = 1.0f - (fmaf(cr, u,  cl * v));   // 1 - A_r
        const float br =         fmaf(cr, uk, cl * vk);   // B_r
        Cl = fmaf(ml, Cl, bl);  Ml *= ml;
        Cr = fmaf(mr, Cr, br);  Mr *= mr;
    }

    // ---- Phase C: wave32 inclusive scan of affine pairs (compose across lanes) ----
#pragma unroll
    for (int d = 1; d < LANES; d <<= 1) {
        const float pMl = __shfl_up(Ml, d, LANES);
        const float pCl = __shfl_up(Cl, d, LANES);
        const float pMr = __shfl_up(Mr, d, LANES);
        const float pCr = __shfl_up(Cr, d, LANES);
        if (lane >= d) {
            Cl = fmaf(Ml, pCl, Cl);  Ml *= pMl;
            Cr = fmaf(Mr, pCr, Cr);  Mr *= pMr;
        }
    }
    // Entry state of lane l = state after lane l-1's chunk (init state is 0).
    float Ql = __shfl_up(Cl, 1, LANES);
    float Qr = __shfl_up(Cr, 1, LANES);
    if (lane == 0) { Ql = 0.0f; Qr = 0.0f; }

    // ---- Phase D: replay 32 trials from LDS, emit (Ql, Qr) per trial ----
    float2* __restrict__ op = (float2*)out + (size_t)sess * N_TRIALS + lane * TPL;
#pragma unroll
    for (int t = 0; t < TPL; ++t) {
        const float cl = my[t * 3 + 0];
        const float cr = my[t * 3 + 1];
        const float o  = my[t * 3 + 2];
        const float u  = fmaf(o, a0 - a1, a1);
        const float v  = fmaf(o, a2 - a3, a3);
        const float uk = fmaf(o, q0 - q1, q1);
        const float vk = fmaf(o, q2 - q3, q3);
        const float ml = 1.0f - (fmaf(cl, u,  cr * v));
        const float bl =         fmaf(cl, uk, cr * vk);
        const float mr = 1.0f - (fmaf(cr, u,  cl * v));
        const float br =         fmaf(cr, uk, cl * vk);
        Ql = fmaf(ml, Ql, bl);
        Qr = fmaf(mr, Qr, br);
        op[t] = make_float2(Ql, Qr);
    }
}

extern "C" void kernel_launch(void* const* d_in, const int* in_sizes, int n_in,
                              void* d_out, int out_size, void* d_ws, size_t ws_size,
                              hipStream_t stream)
{
    const float* inp       = (const float*)d_in[0];  // (8192, 1024, 3)
    const float* alpha_raw = (const float*)d_in[1];  // (4,)
    const float* kvec      = (const float*)d_in[2];  // (4,)
    float*       out       = (float*)d_out;          // (8192, 1024, 2)

    (void)in_sizes; (void)n_in; (void)out_size; (void)d_ws; (void)ws_size;

    const dim3 grid(N_SESS / WAVES_PER_BLOCK);      // 2048 blocks
    const dim3 block(WAVES_PER_BLOCK * LANES);      // 128 threads = 4 wave32
    qvalue_scan_kernel<<<grid, block, 0, stream>>>(inp, alpha_raw, kvec, out);
}